// CE_55937654063537
// MI455X (gfx1250) — compile-verified
//
#include <hip/hip_runtime.h>

typedef __attribute__((ext_vector_type(2))) float v2f;
typedef __attribute__((ext_vector_type(8))) float v8f;

#define H 512
#define W 512
#define PLANE (H * W)          // 262144
#define NPLANES 128            // B(32) * Cin(4)
#define STRIDE 160             // floats of stats per plane
// stats layout per plane:
//  [0..3]   T partials (per quarter)
//  [4..7]   raw row sums, rows 0..3
//  [8..11]  raw row sums, rows 508..511
//  [12..27] left col partials:  col c in 0..3,   slot 12 + c*4 + q
//  [28..43] right col partials: col 508+c,       slot 28 + c*4 + q
//  [44..59] TL 4x4 prefix   [60..75] TR prefix
//  [76..91] BL prefix       [92..107] BR prefix
#define MOFF (NPLANES * STRIDE)   // m buffer: 32*2*49 = 3136 floats

__device__ __forceinline__ float wave_sum(float v) {
    #pragma unroll
    for (int m = 16; m; m >>= 1) v += __shfl_xor(v, m, 32);
    return v;
}

// ---------------- Kernel 1: per-plane boundary/total statistics ----------------
__global__ __launch_bounds__(256) void k_stats(const float* __restrict__ x,
                                               float* __restrict__ stats) {
    const int bid = blockIdx.x;            // 512 blocks
    const int p   = bid >> 2;              // plane (b*4+ci)
    const int q   = bid & 3;               // quarter (128 rows)
    const float* __restrict__ xp = x + (size_t)p * PLANE;
    float* __restrict__ sp = stats + (size_t)p * STRIDE;
    const int tid = threadIdx.x;

    // ---- total-sum partial over 128 rows, float4 streaming ----
    const float4* __restrict__ x4 = (const float4*)xp;
    const int base = q * 16384;            // float4 index of quarter start
    float acc = 0.f;
    #pragma unroll 4
    for (int it = 0; it < 64; ++it) {
        int idx = base + tid + it * 256;
        if (it < 56) __builtin_prefetch(&x4[idx + 2048], 0, 0);  // global_prefetch_b8
        float4 v = x4[idx];
        acc += (v.x + v.y) + (v.z + v.w);
    }
    __shared__ float red[256];
    red[tid] = acc;
    __syncthreads();
    #pragma unroll
    for (int s = 128; s > 0; s >>= 1) {
        if (tid < s) red[tid] += red[tid + s];
        __syncthreads();
    }
    if (tid == 0) sp[q] = red[0];

    const int wid = tid >> 5, lane = tid & 31;

    // ---- boundary column partial sums (this quarter's 128 rows) ----
    {
        int col = (wid < 4) ? wid : 508 + (wid - 4);
        int r0  = q * 128 + lane;
        float c = 0.f;
        #pragma unroll
        for (int k = 0; k < 4; ++k) c += xp[(size_t)(r0 + 32 * k) * W + col];
        c = wave_sum(c);
        if (lane == 0) {
            if (wid < 4) sp[12 + wid * 4 + q]       = c;
            else         sp[28 + (wid - 4) * 4 + q] = c;
        }
    }

    // ---- boundary rows + corner prefixes (only edge quarters) ----
    if (q == 0 || q == 3) {
        if (wid < 4) {
            int row = (q == 0) ? wid : 508 + wid;
            float r = 0.f;
            #pragma unroll
            for (int k = 0; k < 16; ++k) r += xp[(size_t)row * W + lane + 32 * k];
            r = wave_sum(r);
            if (lane == 0) sp[((q == 0) ? 4 : 8) + wid] = r;
        }
        if (tid < 2) {
            float v[4][4];
            #pragma unroll
            for (int r = 0; r < 4; ++r)
                #pragma unroll
                for (int c = 0; c < 4; ++c) {
                    int rr = (q == 0) ? r : 511 - r;
                    int cc = (tid == 0) ? c : 511 - c;
                    v[r][c] = xp[(size_t)rr * W + cc];
                }
            #pragma unroll
            for (int r = 0; r < 4; ++r)
                #pragma unroll
                for (int c = 1; c < 4; ++c) v[r][c] += v[r][c - 1];
            #pragma unroll
            for (int r = 1; r < 4; ++r)
                #pragma unroll
                for (int c = 0; c < 4; ++c) v[r][c] += v[r - 1][c];
            int off = (q == 0) ? ((tid == 0) ? 44 : 60) : ((tid == 0) ? 76 : 92);
            #pragma unroll
            for (int r = 0; r < 4; ++r)
                #pragma unroll
                for (int c = 0; c < 4; ++c) sp[off + r * 4 + c] = v[r][c];
        }
    }
}

// ---------------- Kernel 2: assemble 7x7 maps m = m1 + m2 ----------------
__device__ __forceinline__ float rect_sum(const float* __restrict__ sp,
                                          int nt, int nb, int nl, int nr) {
    // sum over rows [nt .. 511-nb] x cols [nl .. 511-nr] via inclusion-exclusion
    float s = (sp[0] + sp[1]) + (sp[2] + sp[3]);     // T
    for (int k = 0; k < nt; ++k) s -= sp[4 + k];
    for (int k = 0; k < nb; ++k) s -= sp[8 + 3 - k];
    for (int k = 0; k < nl; ++k)
        s -= (sp[12 + k * 4 + 0] + sp[12 + k * 4 + 1]) +
             (sp[12 + k * 4 + 2] + sp[12 + k * 4 + 3]);
    for (int k = 0; k < nr; ++k) {
        int c = 3 - k;
        s -= (sp[28 + c * 4 + 0] + sp[28 + c * 4 + 1]) +
             (sp[28 + c * 4 + 2] + sp[28 + c * 4 + 3]);
    }
    if (nt && nl) s += sp[44 + (nt - 1) * 4 + (nl - 1)];
    if (nt && nr) s += sp[60 + (nt - 1) * 4 + (nr - 1)];
    if (nb && nl) s += sp[76 + (nb - 1) * 4 + (nl - 1)];
    if (nb && nr) s += sp[92 + (nb - 1) * 4 + (nr - 1)];
    return s;
}

__global__ void k_m(const float* __restrict__ stats,
                    const float* __restrict__ g_w, const float* __restrict__ g_b,
                    const float* __restrict__ th_w, const float* __restrict__ th_b,
                    float* __restrict__ mout) {
    int idx = blockIdx.x * blockDim.x + threadIdx.x;
    if (idx >= 32 * 2 * 49) return;
    int j = idx % 7, i = (idx / 7) % 7, c = (idx / 49) & 1, b = idx / 98;

    int nt0 = max(0, i - 3), nb0 = max(0, 3 - i);
    int nl0 = max(0, j - 3), nr0 = max(0, 3 - j);
    float area = (float)((512 - nt0 - nb0) * (512 - nl0 - nr0));
    float acc = (g_b[c] + th_b[c]) * area;

    for (int ci = 0; ci < 4; ++ci) {
        const float* sp = stats + (size_t)(b * 4 + ci) * STRIDE;
        acc += th_w[c * 4 + ci] * rect_sum(sp, nt0, nb0, nl0, nr0);
        #pragma unroll
        for (int dy = -1; dy <= 1; ++dy)
            #pragma unroll
            for (int dx = -1; dx <= 1; ++dx) {
                int nt = max(0, nt0 + dy), nb = max(0, nb0 - dy);
                int nl = max(0, nl0 + dx), nr = max(0, nr0 - dx);
                acc += g_w[((c * 4 + ci) * 3 + (dy + 1)) * 3 + (dx + 1)] *
                       rect_sum(sp, nt, nb, nl, nr);
            }
    }
    mout[idx] = acc * (1.f / 262144.f);
}

// ---------------- Kernel 3: 7x7 -> 512x512 bilinear via WMMA GEMM ----------------
// Out(512x512) = Wy(512x8) * P(8x512),  P = m * Wx^T  (K padded to 8)
__global__ __launch_bounds__(128) void k_up(const float* __restrict__ mbuf,
                                            float* __restrict__ out) {
    __shared__ float Ps[8 * 512];
    __shared__ float ms[49];
    const int bid = blockIdx.x;         // 64 planes * 32 row tiles
    const int p   = bid >> 5;
    const int rt  = bid & 31;
    const int tid = threadIdx.x;
    const float sc = 7.0f / 512.0f;

    if (tid < 49) ms[tid] = mbuf[p * 49 + tid];
    __syncthreads();

    // build P[i][x] = sum_j m[i][j] * wx[x][j]  (bilinear => 2 taps, edge clamp)
    #pragma unroll
    for (int cc = 0; cc < 4; ++cc) {
        int xc = tid * 4 + cc;
        float sx = (xc + 0.5f) * sc - 0.5f;
        sx = fminf(fmaxf(sx, 0.f), 6.f);
        int x0 = (int)sx;
        float fx = sx - (float)x0;
        int x1 = min(x0 + 1, 6);
        #pragma unroll
        for (int i2 = 0; i2 < 7; ++i2)
            Ps[i2 * 512 + xc] = ms[i2 * 7 + x0] * (1.f - fx) + ms[i2 * 7 + x1] * fx;
        Ps[7 * 512 + xc] = 0.f;
    }
    __syncthreads();

    const int wid = tid >> 5, lane = tid & 31;
    const bool lo = lane < 16;
    const int mr  = lane & 15;

    // A operand: 16x8 slice of Wy for rows rt*16 .. rt*16+15
    int y = rt * 16 + mr;
    float sy = ((float)y + 0.5f) * sc - 0.5f;
    sy = fminf(fmaxf(sy, 0.f), 6.f);
    int i0 = (int)sy;
    float fy = sy - (float)i0;
    float wv[8];
    #pragma unroll
    for (int k = 0; k < 8; ++k)
        wv[k] = (k == i0) ? (1.f - fy) : ((k == i0 + 1) ? fy : 0.f);
    v2f a0, a1;
    a0.x = lo ? wv[0] : wv[2];  a0.y = lo ? wv[1] : wv[3];
    a1.x = lo ? wv[4] : wv[6];  a1.y = lo ? wv[5] : wv[7];

    float* __restrict__ outp = out + (size_t)p * PLANE + (size_t)rt * 16 * 512;

    #pragma unroll
    for (int t = 0; t < 8; ++t) {
        int n = wid * 128 + t * 16 + mr;
        v2f b0, b1;
        b0.x = Ps[(lo ? 0 : 2) * 512 + n];  b0.y = Ps[(lo ? 1 : 3) * 512 + n];
        b1.x = Ps[(lo ? 4 : 6) * 512 + n];  b1.y = Ps[(lo ? 5 : 7) * 512 + n];
        v8f cacc = {};
        cacc = __builtin_amdgcn_wmma_f32_16x16x4_f32(false, a0, false, b0,
                                                     (short)0, cacc, false, false);
        cacc = __builtin_amdgcn_wmma_f32_16x16x4_f32(false, a1, false, b1,
                                                     (short)0, cacc, false, false);
        int rowoff = lo ? 0 : 8;
        #pragma unroll
        for (int r = 0; r < 8; ++r)
            outp[(size_t)(r + rowoff) * 512 + n] = cacc[r];
    }
}

extern "C" void kernel_launch(void* const* d_in, const int* in_sizes, int n_in,
                              void* d_out, int out_size, void* d_ws, size_t ws_size,
                              hipStream_t stream) {
    const float* x    = (const float*)d_in[0];
    const float* g_w  = (const float*)d_in[1];
    const float* g_b  = (const float*)d_in[2];
    const float* th_w = (const float*)d_in[3];
    const float* th_b = (const float*)d_in[4];
    float* out   = (float*)d_out;
    float* stats = (float*)d_ws;
    float* mbuf  = stats + MOFF;

    k_stats<<<NPLANES * 4, 256, 0, stream>>>(x, stats);
    k_m<<<(3136 + 255) / 256, 256, 0, stream>>>(stats, g_w, g_b, th_w, th_b, mbuf);
    k_up<<<64 * 32, 128, 0, stream>>>(mbuf, out);
}